// GraphSAGENetwork_17197049053739
// MI455X (gfx1250) — compile-verified
//
#include <hip/hip_runtime.h>

#define N_NODES 50000
#define N_EDGES 1600000
#define DFEAT   128
#define N_GRAPHS 512

typedef __attribute__((ext_vector_type(2))) float v2f;
typedef __attribute__((ext_vector_type(8))) float v8f;

// ---------------------------------------------------------------- utilities
__global__ void fill_zero_kernel(float* __restrict__ p, long n) {
    long i = (long)blockIdx.x * blockDim.x + threadIdx.x;
    long stride = (long)gridDim.x * blockDim.x;
    for (; i < n; i += stride) p[i] = 0.0f;
}

// ------------------------------------------------------------------ degree
__global__ void deg_kernel(const long long* __restrict__ dst,
                           float* __restrict__ deg) {
    int e = blockIdx.x * blockDim.x + threadIdx.x;
    if (e < N_EDGES) atomicAdd(deg + (int)dst[e], 1.0f);
}

__global__ void rdeg_kernel(float* __restrict__ deg) {
    int i = blockIdx.x * blockDim.x + threadIdx.x;
    if (i < N_NODES) deg[i] = 1.0f / fmaxf(deg[i], 1.0f);
}

// ------------------------------------------------- edge gather/scatter-add
// one thread per (edge, float4 chunk): agg[dst][4q..4q+3] += x[src][4q..4q+3]
// b128 gather + 4 scalar f32 atomics; 32 lanes cover one edge's 128 features.
__global__ void scatter_kernel(const float4* __restrict__ xin4,
                               const long long* __restrict__ src,
                               const long long* __restrict__ dst,
                               float* __restrict__ agg) {
    long t = (long)blockIdx.x * blockDim.x + threadIdx.x;
    if (t >= (long)N_EDGES * (DFEAT / 4)) return;
    int e = (int)(t >> 5);          // edge
    int q = (int)(t & 31);          // float4 chunk 0..31
    int s  = (int)src[e];
    int dd = (int)dst[e];
    float4 v = xin4[(long)s * (DFEAT / 4) + q];
    float* base = agg + (long)dd * DFEAT + q * 4;
    atomicAdd(base + 0, v.x);
    atomicAdd(base + 1, v.y);
    atomicAdd(base + 2, v.z);
    atomicAdd(base + 3, v.w);
}

// --------------------------------------------------------- fused SAGE GEMM
// out = relu( (agg * rdeg) @ Wl^T + b + x @ Wr^T )
// one wave32 per 16-row tile; 8 column tiles of 16; K=128 in steps of 4
// using V_WMMA_F32_16X16X4_F32 (full fp32 accumulate).
__global__ __launch_bounds__(256) void sage_gemm_kernel(
        const float* __restrict__ xin,  const float* __restrict__ agg,
        const float* __restrict__ rdeg,
        const float* __restrict__ Wl,   const float* __restrict__ bias,
        const float* __restrict__ Wr,   float* __restrict__ out) {
    int wid  = (blockIdx.x * blockDim.x + threadIdx.x) >> 5;   // wave id (uniform)
    int lane = threadIdx.x & 31;
    if (wid >= N_NODES / 16) return;                            // wave-uniform exit

    int m0    = wid * 16;
    int l15   = lane & 15;
    int hi    = lane >> 4;        // 0: K,K+1  1: K+2,K+3 (A/B frag layout)
    int khalf = hi * 2;
    int row   = m0 + l15;         // A-matrix row handled by this lane
    float rd  = rdeg[row];

    const float* aggRow = agg + (long)row * DFEAT;
    const float* xRow   = xin + (long)row * DFEAT;

    for (int nt = 0; nt < 8; ++nt) {
        int col = nt * 16 + l15;                  // B column / output column
        const float* wlCol = Wl + (long)col * DFEAT;  // Wl^T[k][col] = Wl[col*D + k]
        const float* wrCol = Wr + (long)col * DFEAT;
        float bv = bias[col];
        v8f c = {bv, bv, bv, bv, bv, bv, bv, bv};  // accumulator init = bias

        #pragma unroll 8
        for (int k0 = 0; k0 < DFEAT; k0 += 4) {
            int k = k0 + khalf;                    // even -> 8B aligned pairs
            v2f a1 = *(const v2f*)(aggRow + k);
            a1.x *= rd; a1.y *= rd;                // fold mean-normalization
            v2f b1 = *(const v2f*)(wlCol + k);
            c = __builtin_amdgcn_wmma_f32_16x16x4_f32(
                    false, a1, false, b1, (short)0, c, false, false);
            v2f a2 = *(const v2f*)(xRow + k);
            v2f b2 = *(const v2f*)(wrCol + k);
            c = __builtin_amdgcn_wmma_f32_16x16x4_f32(
                    false, a2, false, b2, (short)0, c, false, false);
        }

        // C/D layout: VGPR r -> M = r (+8 for lanes 16..31), N = lane&15
        int mbase = m0 + hi * 8;
        #pragma unroll
        for (int r = 0; r < 8; ++r) {
            out[(long)(mbase + r) * DFEAT + col] = fmaxf(c[r], 0.0f);
        }
    }
}

// ------------------------------------------------------------ mean pooling
__global__ void pool_accum_kernel(const float4* __restrict__ h4,
                                  const long long* __restrict__ batch,
                                  float* __restrict__ pooled,
                                  float* __restrict__ cnt) {
    long t = (long)blockIdx.x * blockDim.x + threadIdx.x;
    if (t >= (long)N_NODES * (DFEAT / 4)) return;
    int node = (int)(t >> 5);
    int q    = (int)(t & 31);
    int g = (int)batch[node];
    float4 v = h4[(long)node * (DFEAT / 4) + q];
    float* base = pooled + (long)g * DFEAT + q * 4;
    atomicAdd(base + 0, v.x);
    atomicAdd(base + 1, v.y);
    atomicAdd(base + 2, v.z);
    atomicAdd(base + 3, v.w);
    if (q == 0) atomicAdd(cnt + g, 1.0f);
}

__global__ void pool_final_kernel(const float* __restrict__ pooled,
                                  const float* __restrict__ cnt,
                                  const float* __restrict__ fcw,
                                  const float* __restrict__ fcb,
                                  float* __restrict__ outp) {
    int g = blockIdx.x * blockDim.x + threadIdx.x;
    if (g >= N_GRAPHS) return;
    float inv = 1.0f / fmaxf(cnt[g], 1.0f);
    float acc = 0.0f;
    #pragma unroll 4
    for (int d = 0; d < DFEAT; ++d) acc += pooled[g * DFEAT + d] * fcw[d];
    outp[g] = acc * inv + fcb[0];
}

// ------------------------------------------------------------------ launch
extern "C" void kernel_launch(void* const* d_in, const int* in_sizes, int n_in,
                              void* d_out, int out_size, void* d_ws, size_t ws_size,
                              hipStream_t stream) {
    (void)in_sizes; (void)n_in; (void)out_size; (void)ws_size;

    const float*     x     = (const float*)d_in[0];
    const long long* ei    = (const long long*)d_in[1];
    const long long* src   = ei;
    const long long* dst   = ei + N_EDGES;
    const long long* batch = (const long long*)d_in[2];
    const float* Wl[3] = {(const float*)d_in[3], (const float*)d_in[6], (const float*)d_in[9]};
    const float* bs[3] = {(const float*)d_in[4], (const float*)d_in[7], (const float*)d_in[10]};
    const float* Wr[3] = {(const float*)d_in[5], (const float*)d_in[8], (const float*)d_in[11]};
    const float* fcw = (const float*)d_in[12];
    const float* fcb = (const float*)d_in[13];
    float* outp = (float*)d_out;

    // workspace layout (floats): agg | hA | hB | deg | pooled | cnt
    const long NF = (long)N_NODES * DFEAT;
    float* ws     = (float*)d_ws;
    float* agg    = ws;
    float* hA     = ws + NF;
    float* hB     = ws + 2 * NF;
    float* deg    = ws + 3 * NF;
    float* pooled = deg + N_NODES;
    float* cnt    = pooled + (long)N_GRAPHS * DFEAT;

    // degrees (layer-invariant): deg -> 1/max(deg,1)
    fill_zero_kernel<<<512, 256, 0, stream>>>(deg, N_NODES);
    deg_kernel <<<(N_EDGES + 255) / 256, 256, 0, stream>>>(dst, deg);
    rdeg_kernel<<<(N_NODES + 255) / 256, 256, 0, stream>>>(deg);

    const float* in = x;
    float* outs[3] = {hA, hB, hA};
    const long scatterThreads = (long)N_EDGES * (DFEAT / 4);
    const int  scatterBlocks  = (int)((scatterThreads + 255) / 256);
    const int  gemmBlocks     = ((N_NODES / 16) * 32 + 255) / 256;

    for (int l = 0; l < 3; ++l) {
        fill_zero_kernel<<<2048, 256, 0, stream>>>(agg, NF);
        scatter_kernel<<<scatterBlocks, 256, 0, stream>>>(
            (const float4*)in, src, dst, agg);
        sage_gemm_kernel<<<gemmBlocks, 256, 0, stream>>>(
            in, agg, deg, Wl[l], bs[l], Wr[l], outs[l]);
        in = outs[l];
    }

    // global mean pool + fc
    fill_zero_kernel<<<256, 256, 0, stream>>>(pooled, (long)N_GRAPHS * DFEAT + N_GRAPHS);
    const long poolThreads = (long)N_NODES * (DFEAT / 4);
    pool_accum_kernel<<<(int)((poolThreads + 255) / 256), 256, 0, stream>>>(
        (const float4*)in, batch, pooled, cnt);
    pool_final_kernel<<<(N_GRAPHS + 255) / 256, 256, 0, stream>>>(pooled, cnt, fcw, fcb, outp);
}